// GCN_48524540510787
// MI455X (gfx1250) — compile-verified
//
#include <hip/hip_runtime.h>
#include <math.h>

typedef __attribute__((ext_vector_type(2))) float v2f;
typedef __attribute__((ext_vector_type(8))) float v8f;

#define WMMA_F32(a, b, c) \
    __builtin_amdgcn_wmma_f32_16x16x4_f32(false, (a), false, (b), (short)0, (c), false, false)

// ---------------------------------------------------------------- utilities

__global__ __launch_bounds__(256) void fill_kernel(float* __restrict__ p,
                                                   long long n, float v) {
    long long i = (long long)blockIdx.x * blockDim.x + threadIdx.x;
    long long stride = (long long)gridDim.x * blockDim.x;
    for (; i < n; i += stride) p[i] = v;
}

__global__ __launch_bounds__(256) void degree_kernel(float* __restrict__ deg,
                                                     const int* __restrict__ dst,
                                                     int E) {
    int e = blockIdx.x * blockDim.x + threadIdx.x;
    if (e < E) atomicAdd(&deg[dst[e]], 1.0f);
}

__global__ __launch_bounds__(256) void rsqrt_kernel(float* __restrict__ d, int n) {
    int i = blockIdx.x * blockDim.x + threadIdx.x;
    if (i < n) d[i] = rsqrtf(d[i]);   // deg >= 1 thanks to self-loops
}

// ---------------------------------------------------------------- WMMA GEMM
// C[M x F] = A[M x K] @ W[K x F] (+ bias[F] broadcast, optional).
// Requirements: M % 16 == 0, K % 4 == 0, F % 64 == 0 (hold for this problem).
// One wave computes a 32x64 strip = 2 row-tiles x 4 col-tiles of 16x16.
// The 4 B fragments are shared by both row-tiles (12 loads -> 8 WMMAs), and
// the K-loop is software-pipelined: next-step fragments are fetched before the
// current-step WMMAs so loads stay in flight under the matrix ops.
__global__ __launch_bounds__(256) void wmma_gemm_f32(
    const float* __restrict__ A, const float* __restrict__ W,
    const float* __restrict__ bias, float* __restrict__ C,
    int M, int K, int F)
{
    int gtid  = blockIdx.x * blockDim.x + threadIdx.x;
    int wave  = gtid >> 5;
    int lane  = threadIdx.x & 31;
    int strips = F >> 6;                       // F / 64
    int mpair = wave / strips;
    int ntile = wave - mpair * strips;
    int row0  = mpair << 5;                    // 32 rows per wave
    if (row0 >= M) return;                     // wave-uniform
    int col0  = ntile << 6;
    int l15   = lane & 15;
    int hi    = lane >> 4;

    // A row indices; clamp second tile for the M%32 leftover (stores guarded)
    int ra0 = row0 + l15;
    int ra1 = row0 + 16 + l15;
    bool hasHi = (row0 + 16) < M;              // wave-uniform
    if (ra1 >= M) ra1 = ra0;

    v8f c00 = {}, c01 = {}, c02 = {}, c03 = {};
    v8f c10 = {}, c11 = {}, c12 = {}, c13 = {};
    if (bias) {
        float b0 = bias[col0 + l15];
        float b1 = bias[col0 + 16 + l15];
        float b2 = bias[col0 + 32 + l15];
        float b3 = bias[col0 + 48 + l15];
        for (int r = 0; r < 8; ++r) {
            c00[r] = b0; c01[r] = b1; c02[r] = b2; c03[r] = b3;
            c10[r] = b0; c11[r] = b1; c12[r] = b2; c13[r] = b3;
        }
    }

    // Fragment base pointers
    const float* Ap0 = A + (size_t)ra0 * K + (hi << 1);      // a = {A[r][kb], A[r][kb+1]}
    const float* Ap1 = A + (size_t)ra1 * K + (hi << 1);
    const float* Wp  = W + (size_t)(hi << 1) * F + col0 + l15;
    const size_t Wstep = (size_t)4 * F;

    // ---- prologue: fragments for k = 0
    v2f a0c = *(const v2f*)(Ap0);
    v2f a1c = *(const v2f*)(Ap1);
    v2f b0c, b1c, b2c, b3c;
    b0c[0] = Wp[0];  b0c[1] = Wp[F + 0];
    b1c[0] = Wp[16]; b1c[1] = Wp[F + 16];
    b2c[0] = Wp[32]; b2c[1] = Wp[F + 32];
    b3c[0] = Wp[48]; b3c[1] = Wp[F + 48];

    #pragma unroll 2
    for (int k = 4; k < K; k += 4) {
        // ---- prefetch k fragments (in flight under the WMMAs below)
        const float* Wn = Wp + Wstep;
        v2f a0n = *(const v2f*)(Ap0 + k);
        v2f a1n = *(const v2f*)(Ap1 + k);
        v2f b0n, b1n, b2n, b3n;
        b0n[0] = Wn[0];  b0n[1] = Wn[F + 0];
        b1n[0] = Wn[16]; b1n[1] = Wn[F + 16];
        b2n[0] = Wn[32]; b2n[1] = Wn[F + 32];
        b3n[0] = Wn[48]; b3n[1] = Wn[F + 48];

        // ---- compute with current fragments (8 independent WMMAs)
        c00 = WMMA_F32(a0c, b0c, c00);
        c01 = WMMA_F32(a0c, b1c, c01);
        c02 = WMMA_F32(a0c, b2c, c02);
        c03 = WMMA_F32(a0c, b3c, c03);
        c10 = WMMA_F32(a1c, b0c, c10);
        c11 = WMMA_F32(a1c, b1c, c11);
        c12 = WMMA_F32(a1c, b2c, c12);
        c13 = WMMA_F32(a1c, b3c, c13);

        a0c = a0n; a1c = a1n;
        b0c = b0n; b1c = b1n; b2c = b2n; b3c = b3n;
        Wp = Wn;
    }
    // ---- epilogue: last K-step
    c00 = WMMA_F32(a0c, b0c, c00);
    c01 = WMMA_F32(a0c, b1c, c01);
    c02 = WMMA_F32(a0c, b2c, c02);
    c03 = WMMA_F32(a0c, b3c, c03);
    c10 = WMMA_F32(a1c, b0c, c10);
    c11 = WMMA_F32(a1c, b1c, c11);
    c12 = WMMA_F32(a1c, b2c, c12);
    c13 = WMMA_F32(a1c, b3c, c13);

    // ---- store: VGPR r -> row r (lanes 0-15) / r+8 (lanes 16-31)
    for (int r = 0; r < 8; ++r) {
        int rr = r + (hi << 3);
        float* Crow = C + (size_t)(row0 + rr) * F + col0 + l15;
        Crow[0]  = c00[r];
        Crow[16] = c01[r];
        Crow[32] = c02[r];
        Crow[48] = c03[r];
        if (hasHi) {
            float* Crow1 = C + (size_t)(row0 + 16 + rr) * F + col0 + l15;
            Crow1[0]  = c10[r];
            Crow1[16] = c11[r];
            Crow1[32] = c12[r];
            Crow1[48] = c13[r];
        }
    }
}

// ---------------------------------------------------------------- scatter-add
// For each edge (incl. self-loops e >= E): Agg[dst] += H[src] * dis[src]*dis[dst]
// One thread per (edge, 4 contiguous features): coalesced float4 gather,
// 4x global_atomic_add_f32 (aggregation buffer is L2-resident).
__global__ __launch_bounds__(256) void scatter_kernel(
    const float* __restrict__ H, float* __restrict__ Agg,
    const int* __restrict__ src, const int* __restrict__ dst,
    const float* __restrict__ dis, int E, int Nn, int F)
{
    int fchunks = F >> 2;
    long long t = (long long)blockIdx.x * blockDim.x + threadIdx.x;
    long long total = (long long)(E + Nn) * fchunks;
    if (t >= total) return;
    int e = (int)(t / fchunks);
    int f = ((int)(t - (long long)e * fchunks)) << 2;
    int s, d;
    if (e < E) { s = src[e]; d = dst[e]; }
    else       { s = d = e - E; }
    float nrm = dis[s] * dis[d];
    float4 v = *(const float4*)(H + (size_t)s * F + f);
    float* out = Agg + (size_t)d * F + f;
    atomicAdd(out + 0, v.x * nrm);
    atomicAdd(out + 1, v.y * nrm);
    atomicAdd(out + 2, v.z * nrm);
    atomicAdd(out + 3, v.w * nrm);
}

// ---------------------------------------------------------------- combines

__global__ __launch_bounds__(256) void combine_elu_kernel(
    const float* __restrict__ agg, const float* __restrict__ bg,
    const float* __restrict__ lin, float* __restrict__ out,
    long long n, int F)
{
    long long i = (long long)blockIdx.x * blockDim.x + threadIdx.x;
    long long stride = (long long)gridDim.x * blockDim.x;
    for (; i < n; i += stride) {
        int c = (int)(i % F);                     // F is a power of two
        float x = agg[i] + bg[c];
        float e = x > 0.0f ? x : expm1f(x);
        out[i] = e + lin[i];
    }
}

__global__ __launch_bounds__(256) void combine_add_kernel(
    const float* __restrict__ agg, const float* __restrict__ bg,
    const float* __restrict__ lin, float* __restrict__ out,
    long long n, int F)
{
    long long i = (long long)blockIdx.x * blockDim.x + threadIdx.x;
    long long stride = (long long)gridDim.x * blockDim.x;
    for (; i < n; i += stride) {
        int c = (int)(i % F);
        out[i] = agg[i] + bg[c] + lin[i];
    }
}

// ---------------------------------------------------------------- launcher

static inline long long cdivll(long long a, long long b) { return (a + b - 1) / b; }

extern "C" void kernel_launch(void* const* d_in, const int* in_sizes, int n_in,
                              void* d_out, int out_size, void* d_ws, size_t ws_size,
                              hipStream_t stream) {
    const float* x    = (const float*)d_in[0];
    const int*   eidx = (const int*)d_in[1];
    const float* Wg1  = (const float*)d_in[2];
    const float* bg1  = (const float*)d_in[3];
    const float* Wg2  = (const float*)d_in[4];
    const float* bg2  = (const float*)d_in[5];
    const float* Wg3  = (const float*)d_in[6];
    const float* bg3  = (const float*)d_in[7];
    const float* Wl1  = (const float*)d_in[8];
    const float* bl1  = (const float*)d_in[9];
    const float* Wl2  = (const float*)d_in[10];
    const float* bl2  = (const float*)d_in[11];
    const float* Wl3  = (const float*)d_in[12];
    const float* bl3  = (const float*)d_in[13];

    const int MID  = in_sizes[3];            // 512
    const int MID2 = in_sizes[5];            // 256
    const int OUT  = in_sizes[7];            // 64
    const int IN   = in_sizes[2] / MID;      // 256
    const int Nn   = in_sizes[0] / IN;       // 50000 (multiple of 16)
    const int E    = in_sizes[1] / 2;        // 800000
    const int* srcIdx = eidx;
    const int* dstIdx = eidx + E;

    // workspace: dis[N] | bufH[N*MID] | bufA[N*MID] | bufL[N*MID]
    float* ws   = (float*)d_ws;
    size_t off  = 0;
    float* dis  = ws;                    off += (size_t)((Nn + 63) & ~63);
    float* bufH = ws + off;              off += (size_t)Nn * MID;
    float* bufA = ws + off;              off += (size_t)Nn * MID;
    float* bufL = ws + off;

    const int BLK = 256;
    const int FILLG = 4096;

    auto gemm = [&](const float* A, const float* W, const float* bias,
                    float* C, int M, int K, int F) {
        long long waves = cdivll(M, 32) * (F / 64);
        long long blocks = cdivll(waves * 32, BLK);
        wmma_gemm_f32<<<(unsigned)blocks, BLK, 0, stream>>>(A, W, bias, C, M, K, F);
    };
    auto scatter = [&](const float* H, float* Agg, int F) {
        long long total = (long long)(E + Nn) * (F >> 2);
        scatter_kernel<<<(unsigned)cdivll(total, BLK), BLK, 0, stream>>>(
            H, Agg, srcIdx, dstIdx, dis, E, Nn, F);
    };

    // ---- normalization coefficients: deg (with self-loops) -> rsqrt
    fill_kernel<<<FILLG, BLK, 0, stream>>>(dis, (long long)Nn, 1.0f);
    degree_kernel<<<(unsigned)cdivll(E, BLK), BLK, 0, stream>>>(dis, dstIdx, E);
    rsqrt_kernel<<<(unsigned)cdivll(Nn, BLK), BLK, 0, stream>>>(dis, Nn);

    // ---- layer 1: h1 = elu(gcn(x,Wg1)+bg1) + x@Wl1 + bl1      (F = MID)
    gemm(x, Wl1, bl1, bufL, Nn, IN, MID);               // lin1 (+bias)
    gemm(x, Wg1, nullptr, bufH, Nn, IN, MID);           // Hg1
    fill_kernel<<<FILLG, BLK, 0, stream>>>(bufA, (long long)Nn * MID, 0.0f);
    scatter(bufH, bufA, MID);
    combine_elu_kernel<<<FILLG, BLK, 0, stream>>>(bufA, bg1, bufL, bufH,
                                                  (long long)Nn * MID, MID);
    float* h1 = bufH;

    // ---- layer 2: h2 = elu(gcn(h1,Wg2)+bg2) + h1@Wl2 + bl2    (F = MID2)
    gemm(h1, Wl2, bl2, bufL, Nn, MID, MID2);            // lin2
    float* Hg2  = bufA;
    float* agg2 = bufA + (size_t)Nn * MID2;
    gemm(h1, Wg2, nullptr, Hg2, Nn, MID, MID2);
    fill_kernel<<<FILLG, BLK, 0, stream>>>(agg2, (long long)Nn * MID2, 0.0f);
    scatter(Hg2, agg2, MID2);
    combine_elu_kernel<<<FILLG, BLK, 0, stream>>>(agg2, bg2, bufL, bufH,
                                                  (long long)Nn * MID2, MID2);
    float* h2 = bufH;

    // ---- layer 3: out = gcn(h2,Wg3)+bg3 + h2@Wl3 + bl3        (F = OUT)
    gemm(h2, Wl3, bl3, bufL, Nn, MID2, OUT);            // lin3
    float* Hg3  = bufA;
    float* agg3 = bufA + (size_t)Nn * OUT;
    gemm(h2, Wg3, nullptr, Hg3, Nn, MID2, OUT);
    fill_kernel<<<FILLG, BLK, 0, stream>>>(agg3, (long long)Nn * OUT, 0.0f);
    scatter(Hg3, agg3, OUT);
    combine_add_kernel<<<FILLG, BLK, 0, stream>>>(agg3, bg3, bufL, (float*)d_out,
                                                  (long long)Nn * OUT, OUT);
}